// STMTorch_44212393345205
// MI455X (gfx1250) — compile-verified
//
#include <hip/hip_runtime.h>

// ---------------------------------------------------------------------------
// Multiresolution hash-grid encode (Instant-NGP style), fused:
//   out[n, 2l+f] = s[l,f] + t[l,f] * (1 - sigmoid(m))
// s: 16-level 3D grid (8-corner trilinear), t: 16-level 4D grid (16-corner),
// m: tiled 128^3 nearest lookup.
// Gather-bound; all tables (136 MB) resident in MI455X's 192 MB L2.
// Per-level per-dim lo/hi index terms are hoisted so each corner is just
// adds/xors + one global_load_b64; NT hints keep streaming traffic out of L2.
// ---------------------------------------------------------------------------

typedef float __attribute__((ext_vector_type(2))) f32x2;
typedef float __attribute__((ext_vector_type(4))) f32x4;

#define N_LEVELS   16
#define T_ST       (1u << 19)          // hashmap size for table_s / table_t
#define MASK_ST    (T_ST - 1u)
#define MASK_M     ((1u << 21) - 1u)   // table_m size 2^21 (= 128^3 exactly)
#define P1         2654435761u         // PRIMES[1]
#define P2         805459861u          // PRIMES[2]
#define P3         3674653429u         // PRIMES[3]

// res[l] = ceil(16 * 1.2^l), computed in double precision (matches numpy)
__constant__ unsigned c_res[N_LEVELS] = {
    16u, 20u, 24u, 28u, 34u, 40u, 48u, 58u,
    69u, 83u, 100u, 119u, 143u, 172u, 206u, 247u};

// reference uses stride ("dense") indexing when prod(res) <= T:
//   3D: r^3 <= 2^19  ->  l <= 8   (69^3 = 328509)
//   4D: r^4 <= 2^19  ->  l <= 2   (24^4 = 331776)
#define DENSE3_MAX_L 8
#define DENSE4_MAX_L 2

__global__ __launch_bounds__(256) void hashgrid_fused_kernel(
    const f32x4* __restrict__ x,      // (N,4)
    const f32x2* __restrict__ tab_s,  // (16, 2^19, 2)
    const f32x2* __restrict__ tab_t,  // (16, 2^19, 2)
    const float* __restrict__ tab_m,  // (1, 2^21, 1)
    f32x2* __restrict__ out,          // (N, 16 pairs)
    int npts)
{
    const int n = blockIdx.x * blockDim.x + threadIdx.x;
    if (n >= npts) return;

    // Streamed read-once input: non-temporal so it never evicts table lines.
    const f32x4 xv = __builtin_nontemporal_load(&x[n]);
    const float xd[4] = {xv.x, xv.y, xv.z, xv.w};

    // ---- m: tiled 128^3, nearest (round-half-even like jnp.rint) ----------
    unsigned cm[3];
#pragma unroll
    for (int d = 0; d < 3; ++d) {
        float p = xd[d] * 127.0f;                   // x * (res - 1)
        float r = fminf(fmaxf(rintf(p), 0.0f), 127.0f);
        cm[d] = (unsigned)r;
    }
    const unsigned mi = (cm[0] + (cm[1] << 7) + (cm[2] << 14)) & MASK_M;
    const float mv = tab_m[mi];                     // hot table: default RT
    const float om = 1.0f / (1.0f + __expf(mv));    // 1 - sigmoid(mv)

    f32x2* __restrict__ op = out + (size_t)n * N_LEVELS;

#pragma unroll 1
    for (int l = 0; l < N_LEVELS; ++l) {
        const unsigned r = c_res[l];                // uniform scalar load
        const float rm1  = (float)r - 1.0f;
        const unsigned rmax = r - 1u;

        // Per-dim floor/frac + the only two possible corner coordinates.
        // x in [0,1] => floor(x*(r-1)) in [0, r-1]; clamp matches reference's
        // min(p0 + offs, r-1) exactly (coords are non-negative).
        float w[4];
        unsigned ulo[4], uhi[4];
#pragma unroll
        for (int d = 0; d < 4; ++d) {
            float p  = xd[d] * rm1;
            float p0 = floorf(p);
            w[d]     = p - p0;
            unsigned u = (unsigned)p0;
            ulo[d] = min(u, rmax);
            uhi[d] = min(u + 1u, rmax);
        }

        const f32x2* __restrict__ ts = tab_s + (size_t)l * T_ST;
        const f32x2* __restrict__ tt = tab_t + (size_t)l * T_ST;

        // ---- s: 8 corner gathers (one batch of global_load_b64) -----------
        f32x2 fs[8];
        if (l <= DENSE3_MAX_L) {
            // strides [1, r, r^2]; idx < r^3 <= 2^19, mask is a no-op
            const unsigned r2 = r * r;
            const unsigned d0[2] = {ulo[0],      uhi[0]};
            const unsigned d1[2] = {ulo[1] * r,  uhi[1] * r};
            const unsigned d2[2] = {ulo[2] * r2, uhi[2] * r2};
#pragma unroll
            for (int k = 0; k < 8; ++k)
                fs[k] = ts[d0[k & 1] + d1[(k >> 1) & 1] + d2[(k >> 2) & 1]];
        } else {
            const unsigned h0[2] = {ulo[0],      uhi[0]};       // PRIMES[0]=1
            const unsigned h1[2] = {ulo[1] * P1, uhi[1] * P1};
            const unsigned h2[2] = {ulo[2] * P2, uhi[2] * P2};
#pragma unroll
            for (int k = 0; k < 8; ++k)
                fs[k] = ts[(h0[k & 1] ^ h1[(k >> 1) & 1] ^ h2[(k >> 2) & 1])
                           & MASK_ST];
        }

        // ---- t: 16 corner gathers (second batch, overlaps s latency) ------
        f32x2 ft[16];
        if (l <= DENSE4_MAX_L) {
            // strides [1, r, r^2, r^3]; idx < r^4 <= 2^19, mask is a no-op
            const unsigned r2 = r * r, r3 = r2 * r;
            const unsigned d0[2] = {ulo[0],      uhi[0]};
            const unsigned d1[2] = {ulo[1] * r,  uhi[1] * r};
            const unsigned d2[2] = {ulo[2] * r2, uhi[2] * r2};
            const unsigned d3[2] = {ulo[3] * r3, uhi[3] * r3};
#pragma unroll
            for (int k = 0; k < 16; ++k)
                ft[k] = tt[d0[k & 1] + d1[(k >> 1) & 1] +
                           d2[(k >> 2) & 1] + d3[(k >> 3) & 1]];
        } else {
            const unsigned h0[2] = {ulo[0],      uhi[0]};
            const unsigned h1[2] = {ulo[1] * P1, uhi[1] * P1};
            const unsigned h2[2] = {ulo[2] * P2, uhi[2] * P2};
            const unsigned h3[2] = {ulo[3] * P3, uhi[3] * P3};
#pragma unroll
            for (int k = 0; k < 16; ++k)
                ft[k] = tt[(h0[k & 1] ^ h1[(k >> 1) & 1] ^
                            h2[(k >> 2) & 1] ^ h3[(k >> 3) & 1]) & MASK_ST];
        }

        // ---- weights + accumulation (runs under the load waits) -----------
        // Hoist pairwise dim0/dim1 products and lo/hi factors for dims 2,3.
        const float w0s[2] = {1.0f - w[0], w[0]};
        const float w1s[2] = {1.0f - w[1], w[1]};
        const float w2s[2] = {1.0f - w[2], w[2]};
        const float w3s[2] = {1.0f - w[3], w[3]};
        const float p01[4] = {w0s[0] * w1s[0], w0s[1] * w1s[0],
                              w0s[0] * w1s[1], w0s[1] * w1s[1]};

        float s0 = 0.0f, s1 = 0.0f, t0 = 0.0f, t1 = 0.0f;
#pragma unroll
        for (int k = 0; k < 16; ++k) {
            const float wt3 = p01[k & 3] * w2s[(k >> 2) & 1];
            if (k < 8) {
                s0 += wt3 * fs[k].x;
                s1 += wt3 * fs[k].y;
            }
            const float wt4 = wt3 * w3s[(k >> 3) & 1];
            t0 += wt4 * ft[k].x;
            t1 += wt4 * ft[k].y;
        }

        // Streamed write-once output: non-temporal store (th:TH_STORE_NT)
        // keeps the 51 MB result from evicting the resident tables in L2.
        f32x2 o;
        o.x = s0 + t0 * om;
        o.y = s1 + t1 * om;
        __builtin_nontemporal_store(o, &op[l]);
    }
}

extern "C" void kernel_launch(void* const* d_in, const int* in_sizes, int n_in,
                              void* d_out, int out_size, void* d_ws, size_t ws_size,
                              hipStream_t stream) {
    const f32x4* x     = (const f32x4*)d_in[0];   // (400000, 4) f32
    const f32x2* tab_s = (const f32x2*)d_in[1];   // (16, 2^19, 2) f32
    const f32x2* tab_t = (const f32x2*)d_in[2];   // (16, 2^19, 2) f32
    const float* tab_m = (const float*)d_in[3];   // (1, 2^21, 1) f32
    f32x2* out = (f32x2*)d_out;                   // (400000, 32) f32

    const int npts = in_sizes[0] / 4;
    dim3 block(256);                              // 8 wave32 waves per block
    dim3 grid((npts + 255) / 256);
    hipLaunchKernelGGL(hashgrid_fused_kernel, grid, block, 0, stream,
                       x, tab_s, tab_t, tab_m, out, npts);
}